// VIF_loss_61924838473764
// MI455X (gfx1250) — compile-verified
//
#include <hip/hip_runtime.h>
#include <math.h>

// ---------------------------------------------------------------------------
// VIF loss, fully fused: vertical 55-tap Gaussian conv (WMMA f32 16x16x4),
// horizontal conv (WMMA), per-pixel VIF stats, block reduction -> partials,
// tiny finalize kernel -> scalar.
// Round 2: branch-free banded-weight fetch (zero-padded table), scalarized
// wave-uniform field selection -> EXEC stays all-1s through WMMA loops.
// ---------------------------------------------------------------------------

typedef __attribute__((ext_vector_type(2))) float v2f;
typedef __attribute__((ext_vector_type(8))) float v8f;

#define IMG_H   640
#define IMG_W   640
#define OUT_H   586          // 640 - 54
#define OUT_W   586
#define WS      55
#define RAD     27
#define TWO_SG2 242.0f       // 2 * (11)^2

#define NTX     37           // ceil(586/16)
#define NTY     37
#define NB      8
#define NBLK    (NTX*NTY*NB) // 10952

// LDS layout (floats)
#define IT_P    80           // input tile pitch (cols 0..69 valid, 70..79 zero)
#define IT_R    72           // input tile rows  (rows 0..69 valid, 70..71 zero)
#define IT_SZ   (IT_R*IT_P)  // 5760 per field-source array
#define G_PAD   16           // sGx[d + 16], d in [-15, 71], zeros outside band
#define G_SZ    128
#define G_OFF   0
#define IN_OFF  G_SZ                        // 128
#define V_P     81           // V pitch: 17m mod 64 distinct -> conflict-free cols
#define V_OFF   (IN_OFF + 3*IT_SZ)          // 128 + 17280 = 17408
#define O_OFF   (V_OFF + 8*16*V_P)          // + 10368    = 27776
#define SMEM_FLOATS (O_OFF + 8*256)         // + 2048     = 29824  (119296 B)

__device__ __forceinline__ void vifstats(float mua, float mub,
                                         float eaa, float ebb, float eab,
                                         float& g, float& vid, float& vind) {
    float s1  = fmaxf(eaa - mua * mua, 0.0f);
    float s2  = fmaxf(ebb - mub * mub, 0.0f);
    float s12 = eab - mua * mub;
    g = s12 / (s1 + 1e-10f);
    float sv = s2 - g * s12;
    if (s1 < 1e-10f) { g = 0.0f; sv = s2;  s1 = 0.0f; }
    if (s2 < 1e-10f) { g = 0.0f; sv = 0.0f; }
    if (g  < 0.0f)   { sv = s2;  g = 0.0f; }
    if (sv <= 1e-10f) sv = 1e-10f;
    vid  = log10f(1.0f + g * g * s1 / (sv + 0.05f));
    vind = log10f(1.0f + s1 / 0.05f);
}

// Vertical banded-matmul pass; PROD decided by a *scalar* branch at the call
// site so the K-loop body is branch-free with full EXEC.
template <bool PROD>
__device__ __forceinline__ void vert_pass(const float* __restrict__ p0,
                                          const float* __restrict__ p1,
                                          const float* __restrict__ sGx,
                                          int lm, int lh, v8f* accV)
{
    for (int K0 = 0; K0 < 72; K0 += 4) {
        const int k0 = K0 + 2 * lh;
        v2f a;                                   // A[m][k] = g[k - m], m = lm
        a.x = sGx[k0 - lm + G_PAD];
        a.y = sGx[k0 - lm + G_PAD + 1];
        const int base0 = k0 * IT_P + lm;
#pragma unroll
        for (int cc = 0; cc < 5; ++cc) {
            int i0 = base0 + 16 * cc;
            float u0 = p0[i0], u1 = p0[i0 + IT_P];
            if (PROD) { u0 *= p1[i0]; u1 *= p1[i0 + IT_P]; }
            v2f bb; bb.x = u0; bb.y = u1;        // B[k][n] = field[K0+k][16cc+n]
            accV[cc] = __builtin_amdgcn_wmma_f32_16x16x4_f32(
                false, a, false, bb, (short)0, accV[cc], false, false);
        }
    }
}

__global__ __launch_bounds__(256) void vif_main(
    const float* __restrict__ img1,
    const float* __restrict__ img2,
    const float* __restrict__ imgF,
    float* __restrict__ part)
{
    extern __shared__ float smem[];
    float* sGx = smem + G_OFF;
    float* sIn = smem + IN_OFF;
    float* sV  = smem + V_OFF;
    float* sO  = smem + O_OFF;

    const int tid  = threadIdx.x;
    const int lane = tid & 31;
    const int lm   = lane & 15;
    const int lh   = lane >> 4;
    // Wave id as a compiler-visible scalar (it is wave-uniform by construction)
    const int wv   = __builtin_amdgcn_readfirstlane(tid >> 5);

    const int bx = blockIdx.x, by = blockIdx.y, b = blockIdx.z;
    const int x0 = bx * 16, y0 = by * 16;

    // ---- Gaussian taps, zero-padded band table: sGx[d+16] = g[d] (0 outside)
    if (tid < G_SZ) {
        float sum = 0.0f;
        for (int j = 0; j < WS; ++j) {
            float x = (float)(j - RAD);
            sum += expf(-x * x / TWO_SG2);
        }
        float w = 0.0f;
        int d = (int)tid - G_PAD;
        if (d >= 0 && d < WS) {
            float x = (float)(d - RAD);
            w = expf(-x * x / TWO_SG2) / sum;
        }
        sGx[tid] = w;
    }

    // ---- Stage input tiles (70x70 valid, zero-padded to 72x80) into LDS
    float* t1 = sIn;
    float* t2 = sIn + IT_SZ;
    float* tf = sIn + 2 * IT_SZ;
    const long bbase = (long)b * IMG_H * IMG_W;
    for (int i = tid; i < IT_SZ; i += 256) {
        int r = i / IT_P, c = i % IT_P;
        int gy = y0 + r, gx = x0 + c;
        bool ok = (r < 70) && (c < 70) && (gy < IMG_H) && (gx < IMG_W);
        float v1 = 0.0f, v2 = 0.0f, vF = 0.0f;
        if (ok) {
            long gi = bbase + (long)gy * IMG_W + gx;
            v1 = img1[gi]; v2 = img2[gi]; vF = imgF[gi];
        }
        t1[i] = v1; t2[i] = v2; tf[i] = vF;
    }
    __syncthreads();

    // ---- Per-wave field source (scalar selection -> uniform control flow)
    const float* p0 = (wv == 0 || wv == 3 || wv == 6) ? t1
                    : (wv == 1 || wv == 4 || wv == 7) ? t2 : tf;
    const float* p1 = (wv >= 6) ? tf : p0;

    // ---- Vertical pass: V[16 rows][80 cols] = banded W (A) x input tile (B)
    v8f accV[5];
#pragma unroll
    for (int cc = 0; cc < 5; ++cc)
        accV[cc] = (v8f){0.f,0.f,0.f,0.f,0.f,0.f,0.f,0.f};

    if (wv >= 3) vert_pass<true >(p0, p1, sGx, lm, lh, accV);
    else         vert_pass<false>(p0, p1, sGx, lm, lh, accV);

    // ---- Spill V (C layout -> plain [row][col]) to per-wave LDS region
    float* sVw = sV + wv * (16 * V_P);
#pragma unroll
    for (int r = 0; r < 8; ++r) {
        int m = r + 8 * lh;
#pragma unroll
        for (int cc = 0; cc < 5; ++cc)
            sVw[m * V_P + 16 * cc + lm] = accV[cc][r];
    }
    __syncthreads();

    // ---- Horizontal pass: H[16][16] = V (A) x banded W (B)
    v8f accH = (v8f){0.f,0.f,0.f,0.f,0.f,0.f,0.f,0.f};
    for (int K0 = 0; K0 < 72; K0 += 4) {
        const int k0 = K0 + 2 * lh;
        v2f a;                         // A[m][k] = V[m][K0+k], m = lm
        a.x = sVw[lm * V_P + k0];
        a.y = sVw[lm * V_P + k0 + 1];
        v2f bb;                        // B[k][n] = g[(K0+k) - n], n = lm
        bb.x = sGx[k0 - lm + G_PAD];
        bb.y = sGx[k0 - lm + G_PAD + 1];
        accH = __builtin_amdgcn_wmma_f32_16x16x4_f32(
            false, a, false, bb, (short)0, accH, false, false);
    }

    // ---- Stage 16x16 field result
    float* sOw = sO + wv * 256;
#pragma unroll
    for (int r = 0; r < 8; ++r) {
        int m = r + 8 * lh;
        sOw[m * 16 + lm] = accH[r];
    }
    __syncthreads();

    // ---- Per-pixel VIF stats (one pixel per thread)
    const int pm = tid >> 4, pn = tid & 15;
    const int gy = y0 + pm, gx = x0 + pn;
    const bool valid = (gy < OUT_H) && (gx < OUT_W);

    float mu1 = sO[0*256 + tid], mu2 = sO[1*256 + tid], muF = sO[2*256 + tid];
    float e11 = sO[3*256 + tid], e22 = sO[4*256 + tid], eff = sO[5*256 + tid];
    float e1f = sO[6*256 + tid], e2f = sO[7*256 + tid];

    float g1, vid1, vind1, g2, vid2, vind2;
    vifstats(mu1, muF, e11, eff, e1f, g1, vid1, vind1);
    vifstats(mu2, muF, e22, eff, e2f, g2, vid2, vind2);
    bool L = vind1 < vind2;
    float zc = valid ? ((L ? g1   : g2  ) + 1e-7f) : 0.0f;
    float mc = valid ? ((L ? vid1 : vid2) + 1e-7f) : 0.0f;

    // ---- Block reduction (reuse sV region), deterministic partial output
    float* red = sV;
    red[tid] = zc; red[256 + tid] = mc;
    __syncthreads();
    for (int s = 128; s > 0; s >>= 1) {
        if (tid < s) {
            red[tid]       += red[tid + s];
            red[256 + tid] += red[256 + tid + s];
        }
        __syncthreads();
    }
    if (tid == 0) {
        int bid = (b * gridDim.y + by) * gridDim.x + bx;
        part[2 * bid]     = red[0];
        part[2 * bid + 1] = red[256];
    }
}

__global__ __launch_bounds__(256) void vif_finalize(
    const float* __restrict__ part, int nb, float* __restrict__ out)
{
    __shared__ double rz[256], rm[256];
    double sz = 0.0, sm = 0.0;
    for (int i = threadIdx.x; i < nb; i += 256) {
        sz += (double)part[2 * i];
        sm += (double)part[2 * i + 1];
    }
    rz[threadIdx.x] = sz; rm[threadIdx.x] = sm;
    __syncthreads();
    for (int s = 128; s > 0; s >>= 1) {
        if (threadIdx.x < s) {
            rz[threadIdx.x] += rz[threadIdx.x + s];
            rm[threadIdx.x] += rm[threadIdx.x + s];
        }
        __syncthreads();
    }
    if (threadIdx.x == 0)
        out[0] = (float)(1.0 - 0.03 * (rz[0] / rm[0]));
}

extern "C" void kernel_launch(void* const* d_in, const int* in_sizes, int n_in,
                              void* d_out, int out_size, void* d_ws, size_t ws_size,
                              hipStream_t stream) {
    (void)in_sizes; (void)n_in; (void)out_size; (void)ws_size;
    const float* img1 = (const float*)d_in[0];
    const float* img2 = (const float*)d_in[1];
    const float* imgF = (const float*)d_in[2];
    float* out  = (float*)d_out;
    float* part = (float*)d_ws;

    const size_t smemBytes = (size_t)SMEM_FLOATS * sizeof(float); // 119296 B
    (void)hipFuncSetAttribute((const void*)vif_main,
                              hipFuncAttributeMaxDynamicSharedMemorySize,
                              (int)smemBytes);

    dim3 grid(NTX, NTY, NB), blk(256);
    vif_main<<<grid, blk, smemBytes, stream>>>(img1, img2, imgF, part);
    vif_finalize<<<dim3(1), dim3(256), 0, stream>>>(part, NBLK, out);
}